// STConv_90855738179678
// MI455X (gfx1250) — compile-verified
//
#include <hip/hip_runtime.h>
#include <hip/hip_bf16.h>
#include <math.h>

// ---------------- problem constants ----------------
#define Bq    8
#define Tq    32
#define Nq    1024
#define CINq  64
#define Hq    128
#define COUTq 128
#define KSq   3
#define Eq    32768
#define T1q   30      // T - KS + 1
#define T2q   28      // T1 - KS + 1
#define EPSq  1e-5f

typedef __attribute__((ext_vector_type(2))) float v2f;
typedef __attribute__((ext_vector_type(8))) float v8f;

__device__ __forceinline__ v8f wmma_f32(v2f a, v2f b, v8f c) {
    // V_WMMA_F32_16X16X4_F32 : A 16x4 f32 (2 VGPR), B 4x16 f32 (2 VGPR), C/D 16x16 f32 (8 VGPR)
    return __builtin_amdgcn_wmma_f32_16x16x4_f32(
        false, a, false, b, (short)0, c, false, false);
}

// =====================================================================
// Fused gated temporal conv:  out = relu(P * sigmoid(Q) + R)
// GEMM  M=(B*Tout*N) x Kdim=(Cin*3) x 128, three weight sets sharing A.
// One wave owns 4 M-tiles x 1 O-tile: per k-step 4 A-loads + 6 B-loads
// + 12 WMMA; all loads immediate-offset off four base pointers.
// 64 consecutive M-rows share (b,t) since N % 64 == 0.
// =====================================================================
template<int CIN, int TIN, int TOUT>
__global__ __launch_bounds__(256) void tc_gated_kernel(
    const float* __restrict__ x,
    const float* __restrict__ w1, const float* __restrict__ b1,
    const float* __restrict__ w2, const float* __restrict__ b2,
    const float* __restrict__ w3, const float* __restrict__ b3,
    float* __restrict__ out)
{
    constexpr int  KSTEPS  = (CIN * KSq) / 4;
    constexpr long TSTRIDE = (long)Nq * CIN;          // +1 temporal tap (floats)
    constexpr int  M       = Bq * TOUT * Nq;

    const int wave   = blockIdx.x * 8 + (threadIdx.x >> 5);
    const int lane   = threadIdx.x & 31;
    const int lr     = lane & 15;
    const int half   = lane >> 4;
    const int mg     = wave >> 3;                     // group of 64 M-rows
    const int o_tile = wave & 7;
    if (mg * 64 >= M) return;                         // whole-wave exit

    const int m0 = mg * 64;
    const int n0 = m0 & (Nq - 1);
    const int bt = m0 >> 10;                          // / Nq
    const int t  = bt % TOUT;
    const int b  = bt / TOUT;
    // single A base pointer; all A loads are immediate offsets from it
    const float* xp = x + ((long)(b * TIN + t) * Nq + n0 + lr) * CIN + half * 2;

    const int o = o_tile * 16 + lr;
    // B-frag base pointers: offset = cinb*3 + kt (+3 for .y) are immediates
    const float* w1p = w1 + (long)o * CIN * KSq + half * 6;
    const float* w2p = w2 + (long)o * CIN * KSq + half * 6;
    const float* w3p = w3 + (long)o * CIN * KSq + half * 6;

    v8f accP[4] = {{}, {}, {}, {}};
    v8f accQ[4] = {{}, {}, {}, {}};
    v8f accR[4] = {{}, {}, {}, {}};

    #pragma unroll 8
    for (int ks = 0; ks < KSTEPS; ++ks) {
        const int k4   = ks * 4;
        const int kt   = k4 / CIN;                    // temporal tap (const after unroll)
        const int cinb = k4 - kt * CIN;

        v2f a[4];
        #pragma unroll
        for (int i = 0; i < 4; ++i)
            a[i] = *(const v2f*)(xp + kt * TSTRIDE + cinb + i * 16 * CIN);

        v2f bf1, bf2, bf3;
        bf1.x = w1p[cinb * 3 + kt]; bf1.y = w1p[cinb * 3 + kt + 3];
        bf2.x = w2p[cinb * 3 + kt]; bf2.y = w2p[cinb * 3 + kt + 3];
        bf3.x = w3p[cinb * 3 + kt]; bf3.y = w3p[cinb * 3 + kt + 3];

        #pragma unroll
        for (int i = 0; i < 4; ++i) {
            accP[i] = wmma_f32(a[i], bf1, accP[i]);
            accQ[i] = wmma_f32(a[i], bf2, accQ[i]);
            accR[i] = wmma_f32(a[i], bf3, accR[i]);
        }
    }

    const float bb1 = b1[o], bb2 = b2[o], bb3 = b3[o];
    #pragma unroll
    for (int i = 0; i < 4; ++i) {
        float* op = out + (long)(m0 + i * 16 + half * 8) * 128 + o;
        #pragma unroll
        for (int v = 0; v < 8; ++v) {
            float p = accP[i][v] + bb1;
            float q = accQ[i][v] + bb2;
            float r = accR[i][v] + bb3;
            float h = fmaxf(p * (1.f / (1.f + expf(-q))) + r, 0.f);
            op[(long)v * 128] = h;
        }
    }
}

// =====================================================================
// Cheb einsum GEMM:  D (+)= A[M x 128] * W[128 x 128]  (+bias) (relu)
// One wave owns 4 M-tiles x 1 O-tile.
// =====================================================================
__global__ __launch_bounds__(256) void cheb_gemm_kernel(
    const float* __restrict__ A, const float* __restrict__ W,
    const float* __restrict__ bias, float* __restrict__ D,
    int M, int accumulate, int do_relu)
{
    const int wave   = blockIdx.x * 8 + (threadIdx.x >> 5);
    const int lane   = threadIdx.x & 31;
    const int lr     = lane & 15;
    const int half   = lane >> 4;
    const int mg     = wave >> 3;
    const int o_tile = wave & 7;
    if (mg * 64 >= M) return;

    const int o = o_tile * 16 + lr;
    const float* ap = A + (long)(mg * 64 + lr) * 128 + half * 2;
    const float* wp = W + o + half * 256;             // half*2 K-rows of 128

    v8f acc[4] = {{}, {}, {}, {}};

    #pragma unroll 8
    for (int ks = 0; ks < 32; ++ks) {
        v2f bf;
        bf.x = wp[ks * 512];                          // K = ks*4 + half*2
        bf.y = wp[ks * 512 + 128];                    // K+1
        #pragma unroll
        for (int i = 0; i < 4; ++i) {
            v2f a = *(const v2f*)(ap + ks * 4 + i * 16 * 128);
            acc[i] = wmma_f32(a, bf, acc[i]);
        }
    }

    const float bo = bias ? bias[o] : 0.f;
    #pragma unroll
    for (int i = 0; i < 4; ++i) {
        float* dp = D + (long)(mg * 64 + i * 16 + half * 8) * 128 + o;
        #pragma unroll
        for (int v = 0; v < 8; ++v) {
            float val = acc[i][v] + bo;
            if (accumulate) val += dp[(long)v * 128];
            if (do_relu)    val = fmaxf(val, 0.f);
            dp[(long)v * 128] = val;
        }
    }
}

// =====================================================================
// Graph normalization + propagation helpers
// =====================================================================
__global__ void deg_kernel(const int* __restrict__ ei, const float* __restrict__ ew,
                           float* __restrict__ deg, int E)
{
    int e = blockIdx.x * blockDim.x + threadIdx.x;
    if (e < E) atomicAdd(&deg[ei[e]], ew[e]);
}

__global__ void dinv_kernel(const float* __restrict__ deg, float* __restrict__ dinv, int Nn)
{
    int n = blockIdx.x * blockDim.x + threadIdx.x;
    if (n < Nn) {
        float d = deg[n];
        dinv[n] = (d > 0.f) ? rsqrtf(d) : 0.f;
    }
}

__global__ void normw_kernel(const int* __restrict__ ei, const float* __restrict__ ew,
                             const float* __restrict__ dinv, float* __restrict__ nw, int E)
{
    int e = blockIdx.x * blockDim.x + threadIdx.x;
    if (e < E) nw[e] = -dinv[ei[e]] * ew[e] * dinv[ei[E + e]];
}

// Zout[bt, dst, h] += coeff * nw[e] * Zin[bt, src, h]
__global__ void scatter_kernel(const int* __restrict__ ei, const float* __restrict__ nw,
                               const float* __restrict__ Zin, float* __restrict__ Zout,
                               float coeff, int E, int BT, int Nn)
{
    int tid = blockIdx.x * blockDim.x + threadIdx.x;
    int e = tid >> 7, h = tid & 127;
    if (e >= E) return;
    int s = ei[e];
    int d = ei[E + e];
    float w = coeff * nw[e];
    const float* zin = Zin + (long)s * 128 + h;
    float* zout      = Zout + (long)d * 128 + h;
    const long stride = (long)Nn * 128;
    for (int bt = 0; bt < BT; ++bt) {
        atomicAdd(zout, w * (*zin));
        zin += stride; zout += stride;
    }
}

__global__ void neg_kernel(float* __restrict__ z, int total)
{
    int i = blockIdx.x * blockDim.x + threadIdx.x;
    if (i < total) z[i] = -z[i];
}

// =====================================================================
// BatchNorm over node axis (training-mode batch stats)
// =====================================================================
__global__ __launch_bounds__(256) void bn_stats_kernel(
    const float* __restrict__ y, float* __restrict__ sums, float* __restrict__ sqs, int Nn)
{
    const int n = blockIdx.x;
    const int cnt = Bq * T2q * 128;
    float s = 0.f, s2 = 0.f;
    for (int j = threadIdx.x; j < cnt; j += blockDim.x) {
        int bt = j >> 7, o = j & 127;
        float v = y[((long)bt * Nn + n) * 128 + o];
        s += v; s2 += v * v;
    }
    __shared__ float sh[256], sh2[256];
    sh[threadIdx.x] = s; sh2[threadIdx.x] = s2;
    __syncthreads();
    for (int off = 128; off > 0; off >>= 1) {
        if (threadIdx.x < off) {
            sh[threadIdx.x]  += sh[threadIdx.x + off];
            sh2[threadIdx.x] += sh2[threadIdx.x + off];
        }
        __syncthreads();
    }
    if (threadIdx.x == 0) { sums[n] = sh[0]; sqs[n] = sh2[0]; }
}

__global__ void bn_apply_kernel(float* __restrict__ y,
                                const float* __restrict__ sums, const float* __restrict__ sqs,
                                const float* __restrict__ gamma, const float* __restrict__ beta,
                                int total, int Nn, float invcnt)
{
    int i = blockIdx.x * blockDim.x + threadIdx.x;
    if (i >= total) return;
    int n = (i >> 7) % Nn;
    float mean = sums[n] * invcnt;
    float var  = sqs[n] * invcnt - mean * mean;
    float v = (y[i] - mean) * rsqrtf(var + EPSq);
    y[i] = v * gamma[n] + beta[n];
}

// =====================================================================
// launch
// =====================================================================
extern "C" void kernel_launch(void* const* d_in, const int* in_sizes, int n_in,
                              void* d_out, int out_size, void* d_ws, size_t ws_size,
                              hipStream_t stream) {
    const float* x      = (const float*)d_in[0];
    const int*   eidx   = (const int*)  d_in[1];
    const float* ew     = (const float*)d_in[2];
    const float* tc1w1  = (const float*)d_in[3];
    const float* tc1b1  = (const float*)d_in[4];
    const float* tc1w2  = (const float*)d_in[5];
    const float* tc1b2  = (const float*)d_in[6];
    const float* tc1w3  = (const float*)d_in[7];
    const float* tc1b3  = (const float*)d_in[8];
    const float* chebW  = (const float*)d_in[9];
    const float* chebB  = (const float*)d_in[10];
    const float* tc2w1  = (const float*)d_in[11];
    const float* tc2b1  = (const float*)d_in[12];
    const float* tc2w2  = (const float*)d_in[13];
    const float* tc2b2  = (const float*)d_in[14];
    const float* tc2w3  = (const float*)d_in[15];
    const float* tc2b3  = (const float*)d_in[16];
    const float* gamma  = (const float*)d_in[17];
    const float* beta   = (const float*)d_in[18];
    float* out = (float*)d_out;

    // ---- workspace layout ----
    const size_t SZ_T = (size_t)(Bq * T1q * Nq) * 128 * sizeof(float);  // 125829120 B
    float* Tx0  = (float*)d_ws;                          // tc1 output / Tx0 (reused as Tx2)
    float* Tx1  = (float*)((char*)d_ws + SZ_T);
    float* Dbuf = (float*)((char*)d_ws + 2 * SZ_T);      // cheb out -> relu -> tc2 input
    float* deg  = (float*)((char*)d_ws + 3 * SZ_T);
    float* dinv = deg + Nq;
    float* nw   = dinv + Nq;
    float* sums = nw + Eq;
    float* sqs  = sums + Nq;

    const int M1  = Bq * T1q * Nq;   // 245760
    const int M2  = Bq * T2q * Nq;   // 229376
    const int BT1 = Bq * T1q;        // 240

    // ---- graph normalization: deg -> dinv -> norm_w ----
    hipMemsetAsync(deg, 0, Nq * sizeof(float), stream);
    deg_kernel<<<(Eq + 255) / 256, 256, 0, stream>>>(eidx, ew, deg, Eq);
    dinv_kernel<<<(Nq + 255) / 256, 256, 0, stream>>>(deg, dinv, Nq);
    normw_kernel<<<(Eq + 255) / 256, 256, 0, stream>>>(eidx, ew, dinv, nw, Eq);

    // ---- temporal conv 1 (fused gated) : x -> Tx0 ----
    tc_gated_kernel<CINq, Tq, T1q><<<M1 / 64, 256, 0, stream>>>(
        x, tc1w1, tc1b1, tc1w2, tc1b2, tc1w3, tc1b3, Tx0);

    // ---- ChebConv K=3 ----
    // pass 0: D = Tx0 * W0 + bias
    cheb_gemm_kernel<<<M1 / 64, 256, 0, stream>>>(Tx0, chebW, chebB, Dbuf, M1, 0, 0);
    // Tx1 = lhat(Tx0)
    hipMemsetAsync(Tx1, 0, SZ_T, stream);
    scatter_kernel<<<(Eq * 128) / 256, 256, 0, stream>>>(eidx, nw, Tx0, Tx1, 1.f, Eq, BT1, Nq);
    // pass 1: D += Tx1 * W1
    cheb_gemm_kernel<<<M1 / 64, 256, 0, stream>>>(Tx1, chebW + Hq * Hq, nullptr, Dbuf, M1, 1, 0);
    // Tx2 = 2*lhat(Tx1) - Tx0   (in place in Tx0)
    neg_kernel<<<(M1 * 128) / 256, 256, 0, stream>>>(Tx0, M1 * 128);
    scatter_kernel<<<(Eq * 128) / 256, 256, 0, stream>>>(eidx, nw, Tx1, Tx0, 2.f, Eq, BT1, Nq);
    // pass 2: D += Tx2 * W2 ; relu
    cheb_gemm_kernel<<<M1 / 64, 256, 0, stream>>>(Tx0, chebW + 2 * Hq * Hq, nullptr, Dbuf, M1, 1, 1);

    // ---- temporal conv 2 (fused gated) : D -> out (raw, pre-BN) ----
    tc_gated_kernel<Hq, T1q, T2q><<<M2 / 64, 256, 0, stream>>>(
        Dbuf, tc2w1, tc2b1, tc2w2, tc2b2, tc2w3, tc2b3, out);

    // ---- BatchNorm over node axis ----
    bn_stats_kernel<<<Nq, 256, 0, stream>>>(out, sums, sqs, Nq);
    const float invcnt = 1.f / (float)(Bq * T2q * COUTq);
    bn_apply_kernel<<<(M2 * 128 + 255) / 256, 256, 0, stream>>>(
        out, sums, sqs, gamma, beta, M2 * 128, Nq, invcnt);
}